// Graph_69277822484549
// MI455X (gfx1250) — compile-verified
//
#include <hip/hip_runtime.h>
#include <hip/hip_bf16.h>
#include <stdint.h>

// ---------------------------------------------------------------------------
// 2-layer GraphSAGE for MI455X (gfx1250, wave32).
//
// Optimization: mean(x_j) @ W_l == mean(x_j @ W_l)  (linearity), so GEMM runs
// first with V_WMMA_F32_16X16X4_F32 and edge aggregation happens in the
// *output* feature dimension (64 floats/edge layer1, 8 floats/edge layer2).
// Scatter-adds land in 25.6MB accumulators that live in the 192MB L2.
// ---------------------------------------------------------------------------

#define NNODES 100000
#define NEDGES 1600000
#define FIN    128
#define FHID   64
#define FOUT   8
#define NBATCH 16384

typedef __attribute__((ext_vector_type(2))) float v2f;
typedef __attribute__((ext_vector_type(8))) float v8f;

__device__ __forceinline__ v8f wmma_f32_16x16x4(v2f a, v2f b, v8f c) {
    // D = A(16x4 f32) * B(4x16 f32) + C(16x16 f32)
    return __builtin_amdgcn_wmma_f32_16x16x4_f32(
        /*neg_a=*/false, a, /*neg_b=*/false, b,
        /*c_mod=*/(short)0, c, /*reuse_a=*/false, /*reuse_b=*/false);
}

// --------------------------- workspace layout (floats) ---------------------
// [0]              deg      N
// [N]              agg1     N*64      } zeroed each call (contiguous)
// [65N]            agg2     N*8       }
// [73N]            y1 / h   N*64      (h overwrites y1 after aggregation)
// [137N]           r1       N*64
// [201N]           y2       N*8
// [209N]           r2       N*8
// total 217N floats ~= 86.8 MB

#define ZERO_FLOATS ((size_t)NNODES * 73)

__global__ void zero_kernel(float* __restrict__ p, size_t n) {
    size_t stride = (size_t)gridDim.x * blockDim.x;
    for (size_t i = (size_t)blockIdx.x * blockDim.x + threadIdx.x; i < n; i += stride)
        p[i] = 0.0f;
}

// GEMM1: y1 = x @ W1_l, r1 = x @ W1_r   ([N,128] @ [128,64])
// One wave per 16x16 output tile; 4 waves/block cover the 4 column tiles.
// 6250 row tiles exactly (100000 = 6250*16) -> EXEC all-ones everywhere.
__global__ void __launch_bounds__(128)
gemm1_kernel(const float* __restrict__ x,
             const float* __restrict__ Wl, const float* __restrict__ Wr,
             float* __restrict__ y1, float* __restrict__ r1) {
    const int lane = threadIdx.x & 31;
    const int wave = threadIdx.x >> 5;        // col tile 0..3
    const int row0 = blockIdx.x * 16;
    const int col0 = wave * 16;
    const int half = lane >> 4;               // 0 -> K{0,1}, 1 -> K{2,3}
    const int t    = lane & 15;               // M for A-frag, N for B/C-frag

    v8f accL = {};
    v8f accR = {};
    const float* arow = x + (size_t)(row0 + t) * FIN;
    const float* blc  = Wl + col0 + t;
    const float* brc  = Wr + col0 + t;

    for (int k = 0; k < FIN; k += 4) {
        const int kr = k + 2 * half;
        v2f a = *(const v2f*)(arow + kr);                 // 8B aligned (kr even)
        v2f bL, bR;
        bL.x = blc[(size_t)kr * FHID];       bL.y = blc[(size_t)(kr + 1) * FHID];
        bR.x = brc[(size_t)kr * FHID];       bR.y = brc[(size_t)(kr + 1) * FHID];
        accL = wmma_f32_16x16x4(a, bL, accL);
        accR = wmma_f32_16x16x4(a, bR, accR);
    }
#pragma unroll
    for (int v = 0; v < 8; ++v) {
        const int node = row0 + v + 8 * half;
        y1[(size_t)node * FHID + col0 + t] = accL[v];
        r1[(size_t)node * FHID + col0 + t] = accR[v];
    }
}

// Edge scatter-add layer 1: agg1[dst] += y1[src]; deg[dst] += 1.
// 16 threads/edge, each owns a float4 chunk of the 64-float message.
__global__ void __launch_bounds__(256)
edge_agg1_kernel(const int* __restrict__ src, const int* __restrict__ dst,
                 const float* __restrict__ y1,
                 float* __restrict__ agg1, float* __restrict__ deg) {
    const long long tid = (long long)blockIdx.x * blockDim.x + threadIdx.x;
    if (tid >= (long long)NEDGES * 16) return;
    const int e = (int)(tid >> 4);
    const int c = ((int)tid & 15) * 4;
    const int s = src[e], d = dst[e];
    const float4 v = *(const float4*)(y1 + (size_t)s * FHID + c);
    float* p = agg1 + (size_t)d * FHID + c;
    atomicAdd(p + 0, v.x);
    atomicAdd(p + 1, v.y);
    atomicAdd(p + 2, v.z);
    atomicAdd(p + 3, v.w);
    if (((int)tid & 15) == 0) atomicAdd(deg + d, 1.0f);
}

// h = dropout(relu(agg1/max(deg,1) + r1 + b1)) * 2 ; written over y1 buffer.
__global__ void __launch_bounds__(256)
combine1_kernel(const float* __restrict__ agg1, const float* __restrict__ r1,
                const float* __restrict__ deg,  const float* __restrict__ b1,
                const float* __restrict__ mask, float* __restrict__ h) {
    const long long tid = (long long)blockIdx.x * blockDim.x + threadIdx.x;
    if (tid >= (long long)NNODES * FHID) return;
    const int n = (int)(tid >> 6);
    const int f = (int)tid & 63;
    const float m = agg1[tid] / fmaxf(deg[n], 1.0f);
    float v = m + r1[tid] + b1[f];
    v = fmaxf(v, 0.0f);
    v *= (mask[tid] > 0.5f) ? 2.0f : 0.0f;   // inverted dropout, p=0.5
    h[tid] = v;
}

// GEMM2: y2 = h @ W2_l, r2 = h @ W2_r   ([N,64] @ [64,8], cols padded 8->16)
// One wave per 16-row tile; 2 waves/block; 3125 blocks -> 6250 tiles exactly.
__global__ void __launch_bounds__(64)
gemm2_kernel(const float* __restrict__ h,
             const float* __restrict__ Wl, const float* __restrict__ Wr,
             float* __restrict__ y2, float* __restrict__ r2) {
    const int lane = threadIdx.x & 31;
    const int wave = threadIdx.x >> 5;
    const int row0 = (blockIdx.x * 2 + wave) * 16;
    const int half = lane >> 4;
    const int t    = lane & 15;
    const bool col_ok = (t < FOUT);

    v8f accL = {};
    v8f accR = {};
    const float* arow = h + (size_t)(row0 + t) * FHID;

    for (int k = 0; k < FHID; k += 4) {
        const int kr = k + 2 * half;
        v2f a = *(const v2f*)(arow + kr);
        v2f bL, bR;   // zero-pad columns 8..15 (select, not branch: EXEC stays full)
        bL.x = col_ok ? Wl[(size_t)kr * FOUT + t]       : 0.0f;
        bL.y = col_ok ? Wl[(size_t)(kr + 1) * FOUT + t] : 0.0f;
        bR.x = col_ok ? Wr[(size_t)kr * FOUT + t]       : 0.0f;
        bR.y = col_ok ? Wr[(size_t)(kr + 1) * FOUT + t] : 0.0f;
        accL = wmma_f32_16x16x4(a, bL, accL);
        accR = wmma_f32_16x16x4(a, bR, accR);
    }
#pragma unroll
    for (int v = 0; v < 8; ++v) {
        if (col_ok) {
            const int node = row0 + v + 8 * half;
            y2[(size_t)node * FOUT + t] = accL[v];
            r2[(size_t)node * FOUT + t] = accR[v];
        }
    }
}

// Edge scatter-add layer 2: agg2[dst] += y2[src]  (8 floats/edge, 2 thr/edge).
__global__ void __launch_bounds__(256)
edge_agg2_kernel(const int* __restrict__ src, const int* __restrict__ dst,
                 const float* __restrict__ y2, float* __restrict__ agg2) {
    const long long tid = (long long)blockIdx.x * blockDim.x + threadIdx.x;
    if (tid >= (long long)NEDGES * 2) return;
    const int e = (int)(tid >> 1);
    const int c = ((int)tid & 1) * 4;
    const int s = src[e], d = dst[e];
    const float4 v = *(const float4*)(y2 + (size_t)s * FOUT + c);
    float* p = agg2 + (size_t)d * FOUT + c;
    atomicAdd(p + 0, v.x);
    atomicAdd(p + 1, v.y);
    atomicAdd(p + 2, v.z);
    atomicAdd(p + 3, v.w);
}

// out[i] = agg2[idx[i]]/max(deg,1) + r2[idx[i]] + b2
__global__ void __launch_bounds__(256)
final_kernel(const int* __restrict__ idx, const float* __restrict__ agg2,
             const float* __restrict__ r2, const float* __restrict__ deg,
             const float* __restrict__ b2, float* __restrict__ out) {
    const int tid = blockIdx.x * blockDim.x + threadIdx.x;
    if (tid >= NBATCH * FOUT) return;
    const int i = tid >> 3;
    const int f = tid & 7;
    const int n = idx[i];
    const size_t o = (size_t)n * FOUT + f;
    out[tid] = agg2[o] / fmaxf(deg[n], 1.0f) + r2[o] + b2[f];
}

extern "C" void kernel_launch(void* const* d_in, const int* in_sizes, int n_in,
                              void* d_out, int out_size, void* d_ws, size_t ws_size,
                              hipStream_t stream) {
    const float* node_feats = (const float*)d_in[0];   // [N,128]
    const int*   edge_index = (const int*)d_in[1];     // [2,E]
    const int*   idx        = (const int*)d_in[2];     // [B]
    const float* mask       = (const float*)d_in[3];   // [N,64]
    const float* W1_l       = (const float*)d_in[4];   // [128,64]
    const float* W1_r       = (const float*)d_in[5];
    const float* b1         = (const float*)d_in[6];   // [64]
    const float* W2_l       = (const float*)d_in[7];   // [64,8]
    const float* W2_r       = (const float*)d_in[8];
    const float* b2         = (const float*)d_in[9];   // [8]
    float*       out        = (float*)d_out;           // [B,8]

    const int* src = edge_index;                // row 0
    const int* dst = edge_index + NEDGES;       // row 1

    float* ws = (float*)d_ws;
    float* deg  = ws;                                   // N
    float* agg1 = ws + (size_t)NNODES;                  // N*64
    float* agg2 = ws + (size_t)NNODES * 65;             // N*8
    float* hy   = ws + (size_t)NNODES * 73;             // N*64 (y1, then h)
    float* r1   = ws + (size_t)NNODES * 137;            // N*64
    float* y2   = ws + (size_t)NNODES * 201;            // N*8
    float* r2   = ws + (size_t)NNODES * 209;            // N*8
    (void)ws_size; (void)in_sizes; (void)n_in; (void)out_size;

    // 1. zero accumulators (deg|agg1|agg2 are contiguous)
    zero_kernel<<<4096, 256, 0, stream>>>(deg, ZERO_FLOATS);

    // 2. layer-1 GEMMs via f32 WMMA: 6250 row tiles x 4 col tiles
    gemm1_kernel<<<NNODES / 16, 128, 0, stream>>>(node_feats, W1_l, W1_r, hy, r1);

    // 3. edge aggregation of y1 (+ degree count)
    {
        const long long work = (long long)NEDGES * 16;
        edge_agg1_kernel<<<(unsigned)((work + 255) / 256), 256, 0, stream>>>(
            src, dst, hy, agg1, deg);
    }

    // 4. mean + bias + relu + dropout -> h (overwrites y1 buffer)
    {
        const long long work = (long long)NNODES * FHID;
        combine1_kernel<<<(unsigned)((work + 255) / 256), 256, 0, stream>>>(
            agg1, r1, deg, b1, mask, hy);
    }

    // 5. layer-2 GEMMs via f32 WMMA (N padded 8->16 with zero columns)
    gemm2_kernel<<<NNODES / 32, 64, 0, stream>>>(hy, W2_l, W2_r, y2, r2);

    // 6. edge aggregation of y2
    {
        const long long work = (long long)NEDGES * 2;
        edge_agg2_kernel<<<(unsigned)((work + 255) / 256), 256, 0, stream>>>(
            src, dst, y2, agg2);
    }

    // 7. gather batch rows and combine
    final_kernel<<<(NBATCH * FOUT + 255) / 256, 256, 0, stream>>>(
        idx, agg2, r2, deg, b2, out);
}